// LMMSEEqualizer_8924942041972
// MI455X (gfx1250) — compile-verified
//
#include <hip/hip_runtime.h>

typedef __attribute__((ext_vector_type(2))) float v2f;
typedef __attribute__((ext_vector_type(8))) float v8f;

#define B_  8
#define R_  2
#define M_  16
#define TS_ 8          // TX*S streams per rx channel
#define T_  14
#define N_  256
#define TD_ 12         // data OFDM symbols (t not in {2,11})
#define D_  (TD_*N_)
#define NRE (B_*R_*TD_*N_)   // 49152 resource elements that produce output

__device__ __forceinline__ constexpr int tri(int i, int j){ return i*(i+1)/2 + j; }

// ---------------------------------------------------------------------------
// Packed-f32 complex MAC primitives: each = 2x v_pk_fma_f32 (VOP3P) instead of
// 4 scalar v_fma_f32, using op_sel half-broadcast/swap + per-half neg.
// v2f = {re, im} in a 64-bit VGPR pair.
// ---------------------------------------------------------------------------

// acc -= x * conj(y)
__device__ __forceinline__ void cmac_sub_xconjy(v2f& acc, v2f x, v2f y) {
  asm("v_pk_fma_f32 %0, %1, %2, %0 op_sel:[0,0,0] op_sel_hi:[1,0,1] neg_lo:[1,0,0] neg_hi:[1,0,0]\n\t"
      "v_pk_fma_f32 %0, %1, %2, %0 op_sel:[1,1,0] op_sel_hi:[0,1,1] neg_lo:[1,0,0] neg_hi:[0,0,0]"
      : "+v"(acc) : "v"(x), "v"(y));
}

// acc -= l * x
__device__ __forceinline__ void cmac_sub_lx(v2f& acc, v2f l, v2f x) {
  asm("v_pk_fma_f32 %0, %1, %2, %0 op_sel:[0,0,0] op_sel_hi:[0,1,1] neg_lo:[1,0,0] neg_hi:[1,0,0]\n\t"
      "v_pk_fma_f32 %0, %1, %2, %0 op_sel:[1,1,0] op_sel_hi:[1,0,1] neg_lo:[0,0,0] neg_hi:[1,0,0]"
      : "+v"(acc) : "v"(l), "v"(x));
}

// acc += conj(h) * z
__device__ __forceinline__ void cmac_add_conjhz(v2f& acc, v2f h, v2f z) {
  asm("v_pk_fma_f32 %0, %1, %2, %0 op_sel:[0,0,0] op_sel_hi:[0,1,1] neg_lo:[0,0,0] neg_hi:[0,0,0]\n\t"
      "v_pk_fma_f32 %0, %1, %2, %0 op_sel:[1,1,0] op_sel_hi:[1,0,1] neg_lo:[0,0,0] neg_hi:[1,0,0]"
      : "+v"(acc) : "v"(h), "v"(z));
}

// ---------------------------------------------------------------------------
// Kernel A: one wave per RE. Build Stot = H*H^H + diag(no + 8*err_var) with
// V_WMMA_F32_16X16X4_F32 (16x16 out, K=4 — exact covariance shape).
// A-operand layout (16x4 f32): lane&15 = row, VGPR v holds K = 2*(lane>>4)+v.
// B-operand layout (4x16 f32): lane&15 = col, VGPR v holds K = 2*(lane>>4)+v.
// For B = H^T these are the SAME per-lane values as A => reuse registers.
// Lower triangle stored as v2f at ws[tri(i,j)*NRE + re] (coalesced in B).
// ---------------------------------------------------------------------------
__global__ __launch_bounds__(256) void cov_wmma(const float* __restrict__ hr,
                                                const float* __restrict__ hi,
                                                const float* __restrict__ ev,
                                                const float* __restrict__ no,
                                                v2f* __restrict__ ws)
{
  const int lane = threadIdx.x & 31;
  const int wave = blockIdx.x * (blockDim.x >> 5) + (threadIdx.x >> 5);
  if (wave >= NRE) return;                       // wave-uniform; EXEC stays all-1s

  const int n  =  wave & (N_ - 1);
  const int td = (wave >> 8) % TD_;
  const int r  = ((wave >> 8) / TD_) % R_;
  const int b  =  wave / (N_ * TD_ * R_);
  const int t  = td + (td >= 2) + (td >= 10);    // skip pilot symbols 2 and 11

  const int m    = lane & 15;
  const int half = lane >> 4;
  const int k0   = half * 2;
  const int KS   = T_ * N_;                      // stride between streams in h
  const long hb  = ((long)((b*R_ + r)*M_ + m) * TS_) * KS + (long)t*N_ + n;

  v2f ar0 = { hr[hb + (long)(k0+0)*KS], hr[hb + (long)(k0+1)*KS] };
  v2f ar1 = { hr[hb + (long)(k0+4)*KS], hr[hb + (long)(k0+5)*KS] };
  v2f ai0 = { hi[hb + (long)(k0+0)*KS], hi[hb + (long)(k0+1)*KS] };
  v2f ai1 = { hi[hb + (long)(k0+4)*KS], hi[hb + (long)(k0+5)*KS] };
  v2f nai0 = -ai0, nai1 = -ai1;                  // f32 WMMA has no A/B NEG modifier

  // Re(S) = Hr*Hr^T + Hi*Hi^T   (K=8 -> 2 chained K=4 WMMAs per product)
  v8f sre = {};
  sre = __builtin_amdgcn_wmma_f32_16x16x4_f32(false, ar0, false, ar0, (short)0, sre, false, false);
  sre = __builtin_amdgcn_wmma_f32_16x16x4_f32(false, ar1, false, ar1, (short)0, sre, false, false);
  sre = __builtin_amdgcn_wmma_f32_16x16x4_f32(false, ai0, false, ai0, (short)0, sre, false, false);
  sre = __builtin_amdgcn_wmma_f32_16x16x4_f32(false, ai1, false, ai1, (short)0, sre, false, false);
  // Im(S) = Hi*Hr^T - Hr*Hi^T
  v8f sim = {};
  sim = __builtin_amdgcn_wmma_f32_16x16x4_f32(false, ai0, false, ar0,  (short)0, sim, false, false);
  sim = __builtin_amdgcn_wmma_f32_16x16x4_f32(false, ai1, false, ar1,  (short)0, sim, false, false);
  sim = __builtin_amdgcn_wmma_f32_16x16x4_f32(false, ar0, false, nai0, (short)0, sim, false, false);
  sim = __builtin_amdgcn_wmma_f32_16x16x4_f32(false, ar1, false, nai1, (short)0, sim, false, false);

  // Diagonal loading: S[m][m] += no[b,r,m] + 8*err_var[t,n]
  // C layout: lane holds column N = lane&15; VGPR v holds row M = v + 8*half.
  const int Ncol = lane & 15;
  const float dadd = no[(b*R_ + r)*M_ + Ncol] + 8.0f * ev[t*N_ + n];

  #pragma unroll
  for (int v = 0; v < 8; ++v) {
    const int Mrow = v + 8*half;
    if (Mrow == Ncol) sre[v] += dadd;
  }

  // Store lower triangle (Hermitian) as {re, im}
  #pragma unroll
  for (int v = 0; v < 8; ++v) {
    const int Mrow = v + 8*half;
    if (Mrow >= Ncol)
      ws[(long)tri(Mrow, Ncol)*NRE + wave] = (v2f){sre[v], sim[v]};
  }
}

// ---------------------------------------------------------------------------
// Kernel B: one lane per RE. Register-resident complex Cholesky of the 16x16
// lower triangle. Key algebra: Stot^{-1} = L^{-H}L^{-1}, so only FORWARD
// solves are needed:
//   uy = L^{-1} y,  u = L^{-1} hd_k
//   gh = ||u||^2 (real),  gy = u^H uy
//   x_hat = gy/gh,  no_eff = 1/gh - 1
// All complex MACs via packed v_pk_fma_f32 (2 ops per complex MAC).
// amdgpu_waves_per_eu(1): allow full VGPR file (no spill) — latency-chained
// ALU kernel where spills cost more than occupancy.
// Diagonal slots of L are reused to hold 1/L[j][j] (diag is never re-read by
// later factor steps, which touch only strictly-lower entries).
// ---------------------------------------------------------------------------
__global__ __launch_bounds__(256)
__attribute__((amdgpu_waves_per_eu(1)))
void lmmse_solve(const float* __restrict__ yr,
                 const float* __restrict__ yi,
                 const float* __restrict__ hr,
                 const float* __restrict__ hi,
                 const v2f* __restrict__ ws,
                 float* __restrict__ out)
{
  const int re = blockIdx.x * blockDim.x + threadIdx.x;
  if (re >= NRE) return;

  const int n  =  re & (N_ - 1);
  const int td = (re >> 8) % TD_;
  const int r  = ((re >> 8) / TD_) % R_;
  const int b  =  re / (N_ * TD_ * R_);
  const int t  = td + (td >= 2) + (td >= 10);

  // Load lower triangle (coalesced: lanes differ only in re)
  v2f L[136];
  #pragma unroll
  for (int p = 0; p < 136; ++p) L[p] = ws[(long)p*NRE + re];

  // In-place complex Cholesky; diag slot .x is overwritten with 1/L[j][j]
  #pragma unroll
  for (int j = 0; j < 16; ++j) {
    v2f dacc = (v2f){0.f, 0.f};
    #pragma unroll
    for (int k = 0; k < j; ++k) {
      const v2f v = L[tri(j,k)];
      dacc += v * v;                              // packed |.|^2 accumulation
    }
    float d = L[tri(j,j)].x - dacc.x - dacc.y;
    d = fmaxf(d, 1e-30f);
    const float inv = 1.0f / sqrtf(d);
    L[tri(j,j)] = (v2f){inv, 0.f};                // stash reciprocal diagonal
    #pragma unroll
    for (int i = j+1; i < 16; ++i) {
      v2f a = L[tri(i,j)];
      #pragma unroll
      for (int k = 0; k < j; ++k)
        cmac_sub_xconjy(a, L[tri(i,k)], L[tri(j,k)]);   // a -= L[i][k]*conj(L[j][k])
      L[tri(i,j)] = a * inv;
    }
  }

  // uy = L^{-1} y  (forward solve only; y element loaded in-recurrence)
  v2f uy[16];
  #pragma unroll
  for (int i = 0; i < 16; ++i) {
    const int idx = (((b*R_ + r)*M_ + i)*T_ + t)*N_ + n;
    v2f a = (v2f){yr[idx], yi[idx]};
    #pragma unroll
    for (int k = 0; k < i; ++k) cmac_sub_lx(a, L[tri(i,k)], uy[k]);
    uy[i] = a * L[tri(i,i)].x;
  }

  // Desired stream k of rx r is column 4*r+k of the 8-stream channel
  for (int k = 0; k < 4; ++k) {           // rolled: bounds code size
    const int kk = 4*r + k;
    v2f u[16];
    v2f g2 = (v2f){0.f, 0.f};             // packed ||u||^2 accumulator
    v2f gy = (v2f){0.f, 0.f};
    #pragma unroll
    for (int i = 0; i < 16; ++i) {
      const long idx = ((long)(((b*R_ + r)*M_ + i)*TS_ + kk)*T_ + t)*N_ + n;
      v2f a = (v2f){hr[idx], hi[idx]};    // hd[i]: consumed in-place, never stored
      #pragma unroll
      for (int p = 0; p < i; ++p) cmac_sub_lx(a, L[tri(i,p)], u[p]);
      u[i] = a * L[tri(i,i)].x;
      g2 += u[i] * u[i];                  // gh += |u_i|^2
      cmac_add_conjhz(gy, u[i], uy[i]);   // gy += conj(u_i)*uy_i
    }
    const float gh  = g2.x + g2.y;        // real by construction
    const float inv = 1.0f / gh;
    const float xre = gy.x * inv;
    const float xim = gy.y * inv;
    const float ne  = inv - 1.0f;         // Re(1/gh) - 1

    const int tx  = 2*r + (k >> 1), s = k & 1;
    const int pos = ((b*4 + tx)*2 + s)*D_ + td*N_ + n;
    out[2*pos]     = xre;                 // complex64 x_hat, interleaved
    out[2*pos + 1] = xim;
    out[2*B_*4*2*D_ + pos] = ne;          // no_eff after x_hat block
  }
}

// ---------------------------------------------------------------------------
extern "C" void kernel_launch(void* const* d_in, const int* in_sizes, int n_in,
                              void* d_out, int out_size, void* d_ws, size_t ws_size,
                              hipStream_t stream) {
  const float* yr = (const float*)d_in[0];
  const float* yi = (const float*)d_in[1];
  const float* hr = (const float*)d_in[2];
  const float* hi = (const float*)d_in[3];
  const float* ev = (const float*)d_in[4];   // err_var, [T,N]
  const float* no = (const float*)d_in[5];   // [B,R,M]
  v2f* ws = (v2f*)d_ws;                      // needs 136*8*NRE = ~51 MB
  float* out = (float*)d_out;

  // Kernel A: one wave per RE, 8 waves per 256-thread block
  cov_wmma<<<dim3(NRE/8), dim3(256), 0, stream>>>(hr, hi, ev, no, ws);
  // Kernel B: one lane per RE
  lmmse_solve<<<dim3(NRE/256), dim3(256), 0, stream>>>(yr, yi, hr, hi, ws, out);
}